// MixedFeedFoward_21663815041809
// MI455X (gfx1250) — compile-verified
//
#include <hip/hip_runtime.h>
#include <hip/hip_bf16.h>

// MixedFeedForward supernet forward for MI455X (gfx1250, wave32).
// Two fused masked GEMMs using V_WMMA_F32_16X16X4_F32 (exact f32 matrix math;
// the problem is HBM-bandwidth bound at ~190MB / 23.3TB/s ~ 8us, so f32 WMMA
// matches any low-precision path on speed and the f32 reference on accuracy).
// A-tiles are staged with CDNA5 async global->LDS DMA (ASYNCcnt), W-tiles go
// through VGPRs so the DARTS mask scale fuses into the ds_store. LDS is
// double-buffered: one barrier per K-chunk.

typedef __attribute__((ext_vector_type(2))) float v2f;
typedef __attribute__((ext_vector_type(4))) float v4f;
typedef __attribute__((ext_vector_type(8))) float v8f;

#define BM   128   // block tile M
#define BN    64   // block tile N
#define BK    32   // block tile K
#define LDST  36   // padded LDS row stride in floats (16B-aligned rows, conflict-free frags)

// m0[h,d] = sum_ie a[ie]*(d<E_ie) * sum_ir s[ir]*(h < E_ie*R_ir)
// E = {512,768,1024}, R = {2,3,4}. Valid for h in [0,4096), d in [0,1024).
// mask(h,0) == bm0[h]; mask(0,d) == bm1[d] (softmax sums to 1).
__device__ __forceinline__ float mask_scale(const float a[3], const float s[3], int h, int d) {
  float r = a[2] * ((h < 2048 ? s[0] : 0.f) + (h < 3072 ? s[1] : 0.f) + s[2]);
  if (d < 768)
    r += a[1] * ((h < 1536 ? s[0] : 0.f) + (h < 2304 ? s[1] : 0.f) + (h < 3072 ? s[2] : 0.f));
  if (d < 512)
    r += a[0] * ((h < 1024 ? s[0] : 0.f) + (h < 1536 ? s[1] : 0.f) + (h < 2048 ? s[2] : 0.f));
  return r;
}

__device__ __forceinline__ void softmax3(const float* __restrict__ p, float o[3]) {
  float x0 = p[0], x1 = p[1], x2 = p[2];
  float m  = fmaxf(x0, fmaxf(x1, x2));
  float e0 = __expf(x0 - m), e1 = __expf(x1 - m), e2 = __expf(x2 - m);
  float inv = 1.f / (e0 + e1 + e2);
  o[0] = e0 * inv; o[1] = e1 * inv; o[2] = e2 * inv;
}

// C[M,N] = A[M,K] * (scale(W))[N,K]^T + bias_eff[N], optional ReLU (LAYER==0).
// LAYER==0: weight element (n,k) scaled by mask(h=n, d=k)  (W0: [H,D])
// LAYER==1: weight element (n,k) scaled by mask(h=k, d=n)  (W1: [D,H])
template<int LAYER>
__global__ __launch_bounds__(256)
void ffn_gemm(const float* __restrict__ A, const float* __restrict__ W,
              const float* __restrict__ bias, const float* __restrict__ arch_e,
              const float* __restrict__ arch_m, float* __restrict__ C,
              int M, int N, int K)
{
  __shared__ float As[2][BM * LDST];   // 2 x 18 KB
  __shared__ float Bs[2][BN * LDST];   // 2 x  9 KB

  const int tid   = threadIdx.x;
  const int lane  = tid & 31;
  const int wave  = tid >> 5;       // 0..7
  const int wm    = wave & 3;       // 4 wave-rows of 32
  const int wn    = wave >> 2;      // 2 wave-cols of 32
  const int lcol  = lane & 15;
  const int lhalf = lane >> 4;      // 0/1

  const int m0 = blockIdx.y * BM;
  const int n0 = blockIdx.x * BN;

  float a3[3], s3[3];
  softmax3(arch_e, a3);
  softmax3(arch_m, s3);

  v8f acc[2][2] = {};

  // global<->LDS staging coordinates: 8 threads per row, float4 each
  const int arow = tid >> 3;          // 0..31 (+ p*32)
  const int acol = (tid & 7) * 4;     // 0,4,...,28

  // --- staging helpers -----------------------------------------------------
  auto issue_asyncA = [&](int kk, int buf) {
#pragma unroll
    for (int p = 0; p < 4; ++p) {
      // wave-relative LDS byte offset = low 32 bits of the flat LDS address
      unsigned ldsOff = (unsigned)(unsigned long long)(uintptr_t)
                        &As[buf][(arow + p * 32) * LDST + acol];
      const float* g = &A[(size_t)(m0 + arow + p * 32) * K + kk + acol];
      asm volatile("global_load_async_to_lds_b128 %0, %1, off"
                   :: "v"(ldsOff), "v"(g) : "memory");
    }
  };
  auto loadB = [&](int kk, v4f breg[2]) {
#pragma unroll
    for (int p = 0; p < 2; ++p)
      breg[p] = *(const v4f*)&W[(size_t)(n0 + arow + p * 32) * K + kk + acol];
  };
  auto scaleStoreB = [&](int kk, int buf, v4f breg[2]) {
#pragma unroll
    for (int p = 0; p < 2; ++p) {
      int ng = n0 + arow + p * 32;
      int kg = kk + acol;
      // region boundaries are multiples of 4 -> one scale per aligned float4
      float sc = mask_scale(a3, s3, LAYER == 0 ? ng : kg, LAYER == 0 ? kg : ng);
      *(v4f*)&Bs[buf][(arow + p * 32) * LDST + acol] = breg[p] * sc;
    }
  };

  // --- pipelined main loop (double-buffered LDS, 1 barrier / K-chunk) ------
  v4f breg[2];
  issue_asyncA(0, 0);
  loadB(0, breg);
  scaleStoreB(0, 0, breg);
  asm volatile("s_wait_asynccnt 0x0" ::: "memory");
  __syncthreads();

  for (int kk = 0; kk < K; kk += BK) {
    const int cur = (kk >> 5) & 1;          // BK == 32
    const bool more = (kk + BK) < K;        // wave-uniform branch (EXEC unchanged)
    if (more) {
      issue_asyncA(kk + BK, cur ^ 1);       // async DMA into the other buffer
      loadB(kk + BK, breg);                 // W tile -> regs (needs VALU scaling)
    }

    // 8 WMMA K-steps of 4; fragment: lane l holds K = v + 2*(l/16) -> one float2
#pragma unroll
    for (int k = 0; k < BK; k += 4) {
      v2f af[2], bf[2];
#pragma unroll
      for (int i = 0; i < 2; ++i)
        af[i] = *(const v2f*)&As[cur][(wm * 32 + i * 16 + lcol) * LDST + k + 2 * lhalf];
#pragma unroll
      for (int j = 0; j < 2; ++j)
        bf[j] = *(const v2f*)&Bs[cur][(wn * 32 + j * 16 + lcol) * LDST + k + 2 * lhalf];
#pragma unroll
      for (int i = 0; i < 2; ++i)
#pragma unroll
        for (int j = 0; j < 2; ++j)
          acc[i][j] = __builtin_amdgcn_wmma_f32_16x16x4_f32(
              false, af[i], false, bf[j], (short)0, acc[i][j], false, false);
    }

    if (more) {
      scaleStoreB(kk + BK, cur ^ 1, breg);
      asm volatile("s_wait_asynccnt 0x0" ::: "memory");
      __syncthreads();   // next buffer filled AND everyone done reading cur
    }
  }

  // epilogue: effective bias + optional ReLU; C/D layout: row = g + 8*(lane/16), col = lane%16
#pragma unroll
  for (int j = 0; j < 2; ++j) {
    int ncol = n0 + wn * 32 + j * 16 + lcol;
    float bs = mask_scale(a3, s3, LAYER == 0 ? ncol : 0, LAYER == 0 ? 0 : ncol);
    float bval = bs * bias[ncol];
#pragma unroll
    for (int i = 0; i < 2; ++i) {
      int mbase = m0 + wm * 32 + i * 16 + 8 * lhalf;
#pragma unroll
      for (int g = 0; g < 8; ++g) {
        float v = acc[i][j][g] + bval;
        if (LAYER == 0) v = fmaxf(v, 0.f);
        C[(size_t)(mbase + g) * N + ncol] = v;
      }
    }
  }
}

extern "C" void kernel_launch(void* const* d_in, const int* in_sizes, int n_in,
                              void* d_out, int out_size, void* d_ws, size_t ws_size,
                              hipStream_t stream) {
  const float* x  = (const float*)d_in[0];   // [8,512,1024] -> [4096,1024]
  const float* w0 = (const float*)d_in[1];   // [4096,1024]
  const float* b0 = (const float*)d_in[2];   // [4096]
  const float* w1 = (const float*)d_in[3];   // [1024,4096]
  const float* b1 = (const float*)d_in[4];   // [1024]
  const float* ae = (const float*)d_in[5];   // [3]
  const float* am = (const float*)d_in[6];   // [3]
  float* out  = (float*)d_out;               // [4096,1024]
  float* hbuf = (float*)d_ws;                // [4096,4096] f32 intermediate (64 MB)

  const int T = 4096, D = 1024, H = 4096;
  dim3 block(256);

  // layer 0: h = relu(X * (m0*W0)^T + bm0*b0)   [T,H]
  dim3 g1(H / BN, T / BM);
  ffn_gemm<0><<<g1, block, 0, stream>>>(x, w0, b0, ae, am, hbuf, T, H, D);

  // layer 1: out = h * (m1*W1)^T + bm1*b1       [T,D]
  dim3 g2(D / BN, T / BM);
  ffn_gemm<1><<<g2, block, 0, stream>>>(hbuf, w1, b1, ae, am, out, T, D, H);
}